// SpectralConv2d_59227599012556
// MI455X (gfx1250) — compile-verified
//
#include <hip/hip_runtime.h>

// ---------------------------------------------------------------------------
// SpectralConv2d via 2D DHT on gfx1250, f32 WMMA (V_WMMA_F32_16X16X4_F32).
// B=16, CIN=COUT=64, S=128, M1=M2=16.
//
// dht2(x)      = C @ x @ C          (C = 128x128 cas matrix, symmetric)
// dht2(flipx)  = Cf @ x @ Cf^T      (Cf[k,n] = C[k,127-n])
// cos(atan2(s,c)) = c * rsqrt(c^2+s^2)   -> fused "cosphase", stored in d_out
// compl_mul2d  = A[p]*0.5(W[p]+W[pn]) + A[pn]*0.5(W[p]-W[pn])  (precombined W)
// idht2: spectrum sparse (rows 0..15 & 112..127, cols 0..15) -> cheap inverse
//
// All WMMA operands are pre-swizzled:
//   A-swizzle:  Asw[(p*128+f)*2+s] = A[f][2p+s]   (lane-contiguous float2)
//   B-swizzle:  Bsw[(p*n_)*...]    = float2(B[2p][n], B[2p+1][n])
// so every fragment load is one coalesced b64.
// Swizzled x is staged in d_out and overwritten by cosphase (barrier-protected).
// ---------------------------------------------------------------------------

typedef float v2f __attribute__((ext_vector_type(2)));
typedef float v8f __attribute__((ext_vector_type(8)));

#define NS   128          // spatial size
#define LSTR 132          // LDS row stride (floats): conflict-free A-frag/D-store

// workspace layout (in floats)
#define OFF_CSWZ    0                   // C   A-swizzled
#define OFF_CFSWZ   16384               // Cf  A-swizzled
#define OFF_CBSW    32768               // C   B-swizzled (pair-interleaved)
#define OFF_CFTBSW  49152               // Cf^T B-swizzled
#define OFF_CORNER  65536               // [2][16][16][16][64] cos corners (x,y,b,i)
#define OFF_WCOMB   589824              // [2][256][2][64][64] W+/W-, B-swizzled
#define OFF_SPEC    4784128             // [2][16][64][16][16] out_dht corners
// total: 5,308,416 floats = 20.25 MB

__device__ __forceinline__ v8f wmma4(v2f a, v2f b, v8f c) {
  // D = A(16x4 f32) * B(4x16 f32) + C(16x16 f32)
  return __builtin_amdgcn_wmma_f32_16x16x4_f32(false, a, false, b, (short)0, c,
                                               false, false);
}

__device__ __forceinline__ float casf(int r) {   // cas(2*pi*r/128), r in [0,128)
  float a = (float)r * (3.14159265358979323846f / 64.0f);
  return __cosf(a) + __sinf(a);
}

// ---------------------------------------------------------------------------
// Kernel 1: build swizzled transform matrices.
// ---------------------------------------------------------------------------
__global__ __launch_bounds__(256) void build_mats(float* __restrict__ Cswz,
                                                  float* __restrict__ Cfswz,
                                                  float* __restrict__ Cbsw,
                                                  float* __restrict__ Cftbsw) {
  int t = blockIdx.x * 256 + threadIdx.x;          // 0..16383
  int f = t >> 7, m = t & 127;
  float cfm  = casf((f * m) & 127);                // C[f][m]
  float cff  = casf((f * (127 - m)) & 127);        // Cf[f][m]  = C[f][127-m]
  float cft  = casf((m * (127 - f)) & 127);        // Cf^T[f][m] = C[m][127-f]
  // A-swizzle: [(m/2)*128 + f]*2 + (m&1)
  int pa = ((m >> 1) * NS + f) * 2 + (m & 1);
  Cswz [pa] = cfm;
  Cfswz[pa] = cff;
  // B-swizzle: [(f/2)*128 + m]*2 + (f&1)
  int pb = ((f >> 1) * NS + m) * 2 + (f & 1);
  Cbsw  [pb] = cfm;
  Cftbsw[pb] = cft;
}

// ---------------------------------------------------------------------------
// Kernel 2: precombine weights per corner/mode (stored B-swizzled over K=i):
//   W+[i][o] = 0.5*(W[p] + W[pn]),  W-[i][o] = 0.5*(W[p] - W[pn])
// ---------------------------------------------------------------------------
__global__ __launch_bounds__(256) void prep_weights(const float* __restrict__ w1,
                                                    const float* __restrict__ w2,
                                                    float* __restrict__ wcomb) {
  int bx = blockIdx.x;                 // 0..511
  int c = bx >> 8, mode = bx & 255;
  int x = mode >> 4, y = mode & 15;
  int xn = (16 - x) & 15, yn = (16 - y) & 15;
  int p0 = x * 16 + y, p1 = xn * 16 + yn;
  const float* w = c ? w2 : w1;        // [i][o][16][16]
  float* outp = wcomb + (size_t)(c * 256 + mode) * 2 * 4096;
  for (int io = threadIdx.x; io < 4096; io += 256) {
    int i = io >> 6, o = io & 63;
    int sw = (((i >> 1) << 6) + o) * 2 + (i & 1);   // B-swizzled position
    float a = w[io * 256 + p0];
    float b = w[io * 256 + p1];
    outp[sw]        = 0.5f * (a + b);
    outp[4096 + sw] = 0.5f * (a - b);
  }
}

// ---------------------------------------------------------------------------
// Kernel 3: stage x into d_out in B-swizzled (row-pair interleaved) layout.
//   xsw[img][(p*128+n)*2+s] = x[img][2p+s][n]
// ---------------------------------------------------------------------------
__global__ __launch_bounds__(256) void xswz(const float* __restrict__ x,
                                            float* __restrict__ io) {
  int img = blockIdx.x, t = threadIdx.x;
  const float* xim = x + (size_t)img * 16384;
  float* iop = io + (size_t)img * 16384;
  int c = t & 127, th = t >> 7;
#pragma unroll 4
  for (int i = 0; i < 32; ++i) {
    int p = i * 2 + th;                       // row pair 0..63
    v2f v; v.x = xim[(2 * p) * NS + c]; v.y = xim[(2 * p + 1) * NS + c];
    *(v2f*)(iop + (size_t)(p * NS + c) * 2) = v;
  }
}

// ---------------------------------------------------------------------------
// Kernel 4: forward DHT per image (block = one 128x128 image, 8 waves).
// Wave w owns output rows [16w,16w+16): LDS traffic is wave-private.
// Reads swizzled x from d_out, writes cosphase back to d_out (one barrier).
// ---------------------------------------------------------------------------
__device__ __forceinline__ void mm128(const float* __restrict__ Asw, bool a_lds,
                                      const float* __restrict__ Bsw,
                                      int fbase, int ln, int hi, v8f acc[8]) {
#pragma unroll 2
  for (int k = 0; k < NS; k += 4) {
    v2f a;
    if (a_lds) a = *(const v2f*)(Asw + (fbase + ln) * LSTR + k + 2 * hi);
    else       a = *(const v2f*)(Asw + (size_t)(((k >> 1) + hi) * NS + fbase + ln) * 2);
    const float* bb = Bsw + (size_t)((((k >> 1) + hi) << 7) + ln) * 2;
#pragma unroll
    for (int j = 0; j < 8; ++j)
      acc[j] = wmma4(a, *(const v2f*)(bb + j * 32), acc[j]);
  }
}

__device__ __forceinline__ void store_tiles(float* __restrict__ T, const v8f acc[8],
                                            int fbase, int ln, int hi) {
#pragma unroll
  for (int j = 0; j < 8; ++j)
#pragma unroll
    for (int v = 0; v < 8; ++v)
      T[(fbase + v + 8 * hi) * LSTR + j * 16 + ln] = acc[j][v];
}

__global__ __launch_bounds__(256) void dht_fwd(float* io,   // xsw in / cosphase out
                                               const float* __restrict__ Cswz,
                                               const float* __restrict__ Cfswz,
                                               const float* __restrict__ Cbsw,
                                               const float* __restrict__ Cftbsw,
                                               float* __restrict__ corner) {
  extern __shared__ float lds[];                 // Ts[128*132] | sins[128*132]
  float* Ts   = lds;
  float* sins = lds + NS * LSTR;
  const int img  = blockIdx.x;                   // b*64 + i
  const int tid  = threadIdx.x;
  const int w    = tid >> 5, lane = tid & 31;
  const int ln   = lane & 15, hi = lane >> 4;
  const int fbase = w * 16;
  float* iop = io + (size_t)img * 16384;

  v8f acc[8];

  // T2 = Cf @ x
#pragma unroll
  for (int j = 0; j < 8; ++j) acc[j] = v8f{};
  mm128(Cfswz, false, iop, fbase, ln, hi, acc);
  store_tiles(Ts, acc, fbase, ln, hi);

  // sin = T2 @ Cf^T -> sins
#pragma unroll
  for (int j = 0; j < 8; ++j) acc[j] = v8f{};
  mm128(Ts, true, Cftbsw, fbase, ln, hi, acc);
  store_tiles(sins, acc, fbase, ln, hi);

  // T1 = C @ x  (wave-private rows of Ts)
#pragma unroll
  for (int j = 0; j < 8; ++j) acc[j] = v8f{};
  mm128(Cswz, false, iop, fbase, ln, hi, acc);
  store_tiles(Ts, acc, fbase, ln, hi);

  // All xsw reads are consumed by the WMMAs above -> safe to overwrite io
  // with cosphase after this barrier.
  __syncthreads();

  // cos = T1 @ C, fused epilogue
#pragma unroll
  for (int j = 0; j < 8; ++j) acc[j] = v8f{};
  mm128(Ts, true, Cbsw, fbase, ln, hi, acc);

  if (w == 0 || w == 7) {                        // corner tiles (cols 0..15)
    int cidx = (w == 0) ? 0 : 1;
    int b = img >> 6, ci = img & 63;
    float* cb = corner + (size_t)cidx * 262144;
#pragma unroll
    for (int v = 0; v < 8; ++v) {
      int xr = v + 8 * hi;                       // row within 16x16 tile
      cb[(size_t)((xr * 16 + ln) * 16 + b) * 64 + ci] = acc[0][v];
    }
  }

#pragma unroll
  for (int j = 0; j < 8; ++j)
#pragma unroll
    for (int v = 0; v < 8; ++v) {
      int row = fbase + v + 8 * hi, col = j * 16 + ln;
      float c = acc[j][v];
      float s = sins[row * LSTR + col];
      iop[row * NS + col] = c * __frsqrt_rn(fmaxf(c * c + s * s, 1e-37f));
    }
}

// ---------------------------------------------------------------------------
// Kernel 5: mode mixing. One wave per (corner,mode):
//   out[b,o] = A[p] @ W+  +  A[pn] @ W-     (M=16(B), N=64(COUT), K=64(CIN))
// ---------------------------------------------------------------------------
__global__ __launch_bounds__(128) void mode_mix(const float* __restrict__ corner,
                                                const float* __restrict__ wcomb,
                                                float* __restrict__ spec) {
  int bx = blockIdx.x;                   // 0..127
  int c = bx >> 6, mgrp = bx & 63;
  int tid = threadIdx.x, wid = tid >> 5, lane = tid & 31;
  int ln = lane & 15, hi = lane >> 4;
  int mode = mgrp * 4 + wid;             // 0..255
  int x = mode >> 4, y = mode & 15;
  int pn = ((16 - x) & 15) * 16 + ((16 - y) & 15);

  const float* Ap = corner + (size_t)c * 262144 + (size_t)mode * 1024;  // [b][i]
  const float* An = corner + (size_t)c * 262144 + (size_t)pn * 1024;
  const float* Wp = wcomb + (size_t)(c * 256 + mode) * 2 * 4096;        // B-swz
  const float* Wm = Wp + 4096;

  v8f acc[4];
#pragma unroll
  for (int j = 0; j < 4; ++j) acc[j] = v8f{};

#pragma unroll 2
  for (int k = 0; k < 64; k += 4) {
    v2f ap = *(const v2f*)(Ap + ln * 64 + k + 2 * hi);
    v2f an = *(const v2f*)(An + ln * 64 + k + 2 * hi);
    const float* bp = Wp + (size_t)((((k >> 1) + hi) << 6) + ln) * 2;
    const float* bm = Wm + (size_t)((((k >> 1) + hi) << 6) + ln) * 2;
#pragma unroll
    for (int j = 0; j < 4; ++j) {
      acc[j] = wmma4(ap, *(const v2f*)(bp + j * 32), acc[j]);
      acc[j] = wmma4(an, *(const v2f*)(bm + j * 32), acc[j]);
    }
  }

  float* sp = spec + (size_t)c * 262144;          // [b][o][x*16+y]
#pragma unroll
  for (int j = 0; j < 4; ++j)
#pragma unroll
    for (int v = 0; v < 8; ++v) {
      int b = v + 8 * hi, o = j * 16 + ln;
      sp[((size_t)(b * 64 + o) << 8) + mode] = acc[j][v];
    }
}

// ---------------------------------------------------------------------------
// Kernel 6: sparse inverse DHT + multiply by cosphase (RMW d_out).
//   U(128x16) = Ctop(128x16)@Stop + Cbot(128x16)@Sbot ;  y = (U @ C)/16384
// ---------------------------------------------------------------------------
__global__ __launch_bounds__(256) void inv_dht(const float* __restrict__ Cswz,
                                               const float* __restrict__ Cbsw,
                                               const float* __restrict__ spec,
                                               float* __restrict__ out) {
  __shared__ float Stop[256], Sbot[256], U[NS * 20];   // S* stored B-swizzled
  int img = blockIdx.x, tid = threadIdx.x;
  {
    int m = tid >> 4, n = tid & 15;
    int sw = ((m >> 1) * 16 + n) * 2 + (m & 1);
    Stop[sw] = spec[(size_t)img * 256 + tid];
    Sbot[sw] = spec[262144 + (size_t)img * 256 + tid];
  }
  __syncthreads();

  int w = tid >> 5, lane = tid & 31, ln = lane & 15, hi = lane >> 4;
  int fbase = w * 16;

  v8f u = v8f{};
#pragma unroll
  for (int k = 0; k < 16; k += 4) {              // top rows: C cols 0..15
    v2f a = *(const v2f*)(Cswz + (size_t)(((k >> 1) + hi) * NS + fbase + ln) * 2);
    v2f b = *(const v2f*)(Stop + (((k >> 1) + hi) * 16 + ln) * 2);
    u = wmma4(a, b, u);
  }
#pragma unroll
  for (int k = 0; k < 16; k += 4) {              // bottom rows: C cols 112..127
    v2f a = *(const v2f*)(Cswz + (size_t)((56 + (k >> 1) + hi) * NS + fbase + ln) * 2);
    v2f b = *(const v2f*)(Sbot + (((k >> 1) + hi) * 16 + ln) * 2);
    u = wmma4(a, b, u);
  }
#pragma unroll
  for (int v = 0; v < 8; ++v)
    U[(fbase + v + 8 * hi) * 20 + ln] = u[v];    // wave-private rows, no barrier

  v8f acc[8];
#pragma unroll
  for (int j = 0; j < 8; ++j) acc[j] = v8f{};
#pragma unroll
  for (int k = 0; k < 16; k += 4) {
    v2f a = *(const v2f*)&U[(fbase + ln) * 20 + k + 2 * hi];
    const float* bb = Cbsw + (size_t)((((k >> 1) + hi) << 7) + ln) * 2;
#pragma unroll
    for (int j = 0; j < 8; ++j)
      acc[j] = wmma4(a, *(const v2f*)(bb + j * 32), acc[j]);
  }

  float* op = out + (size_t)img * 16384;
  const float scale = 1.0f / 16384.0f;
#pragma unroll
  for (int j = 0; j < 8; ++j)
#pragma unroll
    for (int v = 0; v < 8; ++v) {
      int row = fbase + v + 8 * hi, col = j * 16 + ln;
      float cp = op[row * NS + col];             // cosphase (same-thread RMW)
      op[row * NS + col] = acc[j][v] * scale * cp;
    }
}

// ---------------------------------------------------------------------------
extern "C" void kernel_launch(void* const* d_in, const int* in_sizes, int n_in,
                              void* d_out, int out_size, void* d_ws, size_t ws_size,
                              hipStream_t stream) {
  const float* x  = (const float*)d_in[0];
  const float* w1 = (const float*)d_in[1];
  const float* w2 = (const float*)d_in[2];
  float* out = (float*)d_out;
  float* ws  = (float*)d_ws;

  float* Cswz   = ws + OFF_CSWZ;
  float* Cfswz  = ws + OFF_CFSWZ;
  float* Cbsw   = ws + OFF_CBSW;
  float* Cftbsw = ws + OFF_CFTBSW;
  float* corner = ws + OFF_CORNER;
  float* wcomb  = ws + OFF_WCOMB;
  float* spec   = ws + OFF_SPEC;

  build_mats<<<64, 256, 0, stream>>>(Cswz, Cfswz, Cbsw, Cftbsw);
  prep_weights<<<512, 256, 0, stream>>>(w1, w2, wcomb);
  xswz<<<1024, 256, 0, stream>>>(x, out);
  dht_fwd<<<1024, 256, 2 * NS * LSTR * sizeof(float), stream>>>(
      out, Cswz, Cfswz, Cbsw, Cftbsw, corner);
  mode_mix<<<128, 128, 0, stream>>>(corner, wcomb, spec);
  inv_dht<<<1024, 256, 0, stream>>>(Cswz, Cbsw, spec, out);
}